// StochasticLayerGCN_79671643341633
// MI455X (gfx1250) — compile-verified
//
#include <hip/hip_runtime.h>

typedef float v2f __attribute__((ext_vector_type(2)));
typedef float v8f __attribute__((ext_vector_type(8)));

// Hardware fp32 atomic add (global_atomic_add_f32, no CAS loop).
__device__ __forceinline__ void fadd_atomic(float* p, float v) {
    unsafeAtomicAdd(p, v);
}

// ---------------------------------------------------------------------------
// Zero a 32-bit region (grid-stride).
// ---------------------------------------------------------------------------
__global__ void gcn_zero(unsigned* __restrict__ p, size_t n) {
    size_t i = (size_t)blockIdx.x * blockDim.x + threadIdx.x;
    size_t stride = (size_t)gridDim.x * blockDim.x;
    for (; i < n; i += stride) p[i] = 0u;
}

// ---------------------------------------------------------------------------
// Integer degree accumulation: degS[src[e]] += 1, degD[dst[e]] += 1.
// Exact counts via global_atomic_add_u32.
// ---------------------------------------------------------------------------
__global__ void gcn_degrees(const int* __restrict__ src, const int* __restrict__ dst,
                            unsigned* __restrict__ degS, unsigned* __restrict__ degD,
                            int E) {
    int e = blockIdx.x * blockDim.x + threadIdx.x;
    if (e >= E) return;
    atomicAdd(degS + src[e], 1u);
    atomicAdd(degD + dst[e], 1u);
}

// ---------------------------------------------------------------------------
// In-place: uint count -> float (count>0 ? count^-1/2 : 0), stored as bits.
// ---------------------------------------------------------------------------
__global__ void gcn_deg_to_rsqrt(unsigned* __restrict__ p, int n) {
    int i = blockIdx.x * blockDim.x + threadIdx.x;
    if (i >= n) return;
    unsigned c = p[i];
    float r = (c > 0u) ? rsqrtf((float)c) : 0.0f;
    p[i] = __float_as_uint(r);
}

// ---------------------------------------------------------------------------
// Edge scatter-add: one wave32 per edge.
//   agg[dst[e], :] += x[src[e], :] * cs[src[e]]
// Each lane handles float4 chunks (F = 128 -> 1 chunk/lane, F = 256 -> 2).
// ---------------------------------------------------------------------------
__global__ void gcn_scatter(const float* __restrict__ x, const int* __restrict__ src,
                            const int* __restrict__ dst, const float* __restrict__ cs,
                            float* __restrict__ agg, int E, int F) {
    int gid = blockIdx.x * blockDim.x + threadIdx.x;
    int e = gid >> 5;
    int lane = gid & 31;
    if (e >= E) return;
    int s = src[e];
    int d = dst[e];
    float c = cs[s];
    const float* xr = x + (size_t)s * F;
    float* ar = agg + (size_t)d * F;
    for (int j = lane * 4; j < F; j += 128) {
        float4 v = *(const float4*)(xr + j);
        fadd_atomic(ar + j + 0, v.x * c);
        fadd_atomic(ar + j + 1, v.y * c);
        fadd_atomic(ar + j + 2, v.z * c);
        fadd_atomic(ar + j + 3, v.w * c);
    }
}

// ---------------------------------------------------------------------------
// out = relu( (A * cd[:,None]) @ W + b ), fp32 WMMA 16x16x4.
// One wave per 16x16 output tile; K stepped by 4.
//
// Fragment layouts (ISA 7.12.2, 32-bit, wave32):
//   A 16x4 : lane<16 -> M=lane,   v[0]=K0,v[1]=K1 ; lane>=16 -> M=lane-16, K2/K3
//   B 4x16 : lane<16 -> N=lane,   v[0]=K0,v[1]=K1 ; lane>=16 -> N=lane-16, K2/K3
//   C/D    : col = lane%16, row = vgprIdx + 8*(lane/16)
// ---------------------------------------------------------------------------
__global__ void gcn_gemm_wmma(const float* __restrict__ A, const float* __restrict__ W,
                              const float* __restrict__ bias, const float* __restrict__ cd,
                              float* __restrict__ out, int M, int K, int N) {
    const int wavesPerBlock = blockDim.x >> 5;
    const int wave = threadIdx.x >> 5;
    const int lane = threadIdx.x & 31;
    const int tilesN = N >> 4;
    const int tilesM = (M + 15) >> 4;
    const int tile = blockIdx.x * wavesPerBlock + wave;
    if (tile >= tilesM * tilesN) return;   // wave-uniform exit: EXEC stays full

    const int mT = tile / tilesN;
    const int nT = tile % tilesN;
    const int hl = lane >> 4;      // 0: K {0,1}, 1: K {2,3}
    const int r  = lane & 15;

    const int m = mT * 16 + r;
    const int n = nT * 16 + r;
    const bool mOK = (m < M);
    const float scale = mOK ? cd[m] : 0.0f;                 // fold cd into A load
    const float* arow = A + (size_t)(mOK ? m : 0) * K;      // clamp: no OOB read

    v8f acc = {};
    for (int kk = 0; kk < K; kk += 4) {
        const int k0 = kk + 2 * hl;
        v2f a, b;
        a.x = arow[k0] * scale;
        a.y = arow[k0 + 1] * scale;
        b.x = W[(size_t)k0 * N + n];          // coalesced across lanes (n contiguous)
        b.y = W[(size_t)(k0 + 1) * N + n];
        acc = __builtin_amdgcn_wmma_f32_16x16x4_f32(
            /*neg_a=*/false, a, /*neg_b=*/false, b,
            /*c_mod=*/(short)0, acc, /*reuse_a=*/false, /*reuse_b=*/false);
    }

    const float bn = bias[n];
    #pragma unroll
    for (int v = 0; v < 8; ++v) {
        const int row = mT * 16 + v + 8 * hl;
        if (row < M) {
            float val = acc[v] + bn;
            out[(size_t)row * N + n] = fmaxf(val, 0.0f);
        }
    }
}

// ---------------------------------------------------------------------------
// Host launcher. Inputs: x, src0, dst0, src1, dst1, W0, b0, W1, b1.
// ---------------------------------------------------------------------------
extern "C" void kernel_launch(void* const* d_in, const int* in_sizes, int n_in,
                              void* d_out, int out_size, void* d_ws, size_t ws_size,
                              hipStream_t stream) {
    const float* x    = (const float*)d_in[0];
    const int*   src0 = (const int*)  d_in[1];
    const int*   dst0 = (const int*)  d_in[2];
    const int*   src1 = (const int*)  d_in[3];
    const int*   dst1 = (const int*)  d_in[4];
    const float* W0   = (const float*)d_in[5];
    const float* b0   = (const float*)d_in[6];
    const float* W1   = (const float*)d_in[7];
    const float* b1   = (const float*)d_in[8];
    float* out = (float*)d_out;

    const int E0 = in_sizes[1];
    const int E1 = in_sizes[3];
    const int NS0 = 400000;    // x rows
    const int ND0 = 100000;    // layer-0 dst nodes
    const int ND1 = 25000;     // layer-1 dst nodes
    const int F0 = 128, H0 = 256, F1 = 256, H1 = 64;

    // Workspace layout (32-bit words). Degree arrays are contiguous so one
    // count->rsqrt pass covers all of them; degrees are uint during
    // accumulation and hold float bits afterwards.
    float* ws   = (float*)d_ws;
    float* cs0  = ws;                              // [NS0]
    float* cd0  = cs0 + NS0;                       // [ND0]
    float* cs1  = cd0 + ND0;                       // [ND0]
    float* cd1  = cs1 + ND0;                       // [ND1]
    float* agg0 = cd1 + ND1;                       // [ND0 * F0]
    float* h0   = agg0 + (size_t)ND0 * F0;         // [ND0 * H0]
    float* agg1 = h0 + (size_t)ND0 * H0;           // [ND1 * F1]

    const size_t nDeg   = (size_t)NS0 + ND0 + ND0 + ND1;         // 625,000
    const size_t nZero0 = nDeg + (size_t)ND0 * F0;               // degs + agg0
    const size_t nZero1 = (size_t)ND1 * F1;                      // agg1

    // 1) Zero accumulators + degree arrays.
    gcn_zero<<<4096, 256, 0, stream>>>((unsigned*)ws, nZero0);
    gcn_zero<<<4096, 256, 0, stream>>>((unsigned*)agg1, nZero1);

    // 2) Integer degree counts for both layers (data-independent).
    gcn_degrees<<<(E0 + 255) / 256, 256, 0, stream>>>(src0, dst0,
                                                      (unsigned*)cs0, (unsigned*)cd0, E0);
    gcn_degrees<<<(E1 + 255) / 256, 256, 0, stream>>>(src1, dst1,
                                                      (unsigned*)cs1, (unsigned*)cd1, E1);

    // 3) count -> count^-1/2 (single pass over the contiguous degree region).
    gcn_deg_to_rsqrt<<<((int)nDeg + 255) / 256, 256, 0, stream>>>((unsigned*)ws, (int)nDeg);

    // 4) Layer 0 aggregation: agg0[d] += x[s] * cs0[s].
    {
        size_t threads = (size_t)E0 * 32;
        gcn_scatter<<<(unsigned)((threads + 255) / 256), 256, 0, stream>>>(
            x, src0, dst0, cs0, agg0, E0, F0);
    }

    // 5) Layer 0 GEMM + bias + ReLU: h0 = relu((agg0*cd0) @ W0 + b0).
    {
        int tiles = ((ND0 + 15) / 16) * (H0 / 16);
        gcn_gemm_wmma<<<(tiles + 7) / 8, 256, 0, stream>>>(
            agg0, W0, b0, cd0, h0, ND0, F0, H0);
    }

    // 6) Layer 1 aggregation: agg1[d] += h0[s] * cs1[s].
    {
        size_t threads = (size_t)E1 * 32;
        gcn_scatter<<<(unsigned)((threads + 255) / 256), 256, 0, stream>>>(
            h0, src1, dst1, cs1, agg1, E1, F1);
    }

    // 7) Layer 1 GEMM + bias + ReLU: out = relu((agg1*cd1) @ W1 + b1).
    {
        int tiles = ((ND1 + 15) / 16) * (H1 / 16);
        gcn_gemm_wmma<<<(tiles + 7) / 8, 256, 0, stream>>>(
            agg1, W1, b1, cd1, out, ND1, F1, H1);
    }
}